// Net_57397942943775
// MI455X (gfx1250) — compile-verified
//
#include <hip/hip_runtime.h>
#include <hip/hip_bf16.h>
#include <math.h>

typedef __attribute__((ext_vector_type(2))) float v2f;
typedef __attribute__((ext_vector_type(4))) float v4f;
typedef __attribute__((ext_vector_type(8))) float v8f;

#define N_NODES   100000
#define N_EDGES   3200000
#define N_FEAT    128
#define DIM       32
#define N_GRAPHS  1000
#define N_CLASSES 2
#define BN_EPS    1e-5f

#define M_TILES   ((N_NODES + 15) / 16)   // 6250
#define WAVES_PER_BLOCK 8

// ---------------------------------------------------------------------------
// Kernel 1: Y[M x 32] = Hin[M x din] @ W[din x 32]   (fp32 WMMA 16x16x4)
// One wave computes a 16x32 output tile (two 16-wide N tiles).
// A frag (16x4 f32): lane<16 row=lane K={k0,k0+1}; lane>=16 row=lane-16 K={k0+2,k0+3}
// B frag (4x16 f32): mirrored K striping, col = lane&15 (+16 for second N tile)
// C/D (16x16 f32):   VGPR v -> row v (lanes 0-15) / row v+8 (lanes 16-31)
// ---------------------------------------------------------------------------
__global__ __launch_bounds__(32 * WAVES_PER_BLOCK)
void gin_lin1_wmma(const float* __restrict__ A, const float* __restrict__ W,
                   float* __restrict__ Y, int din) {
  const int wave = threadIdx.x >> 5;
  const int tile = blockIdx.x * WAVES_PER_BLOCK + wave;
  if (tile >= M_TILES) return;                 // wave-uniform: EXEC all-1s below
  const int lane = threadIdx.x & 31;
  const int half = lane >> 4;                  // 0 or 1
  const int l15  = lane & 15;
  const int m0   = tile * 16;

  v8f c0 = {}; v8f c1 = {};
  const float* arow = A + (size_t)(m0 + l15) * din + half * 2;

  #pragma unroll 4
  for (int k0 = 0; k0 < din; k0 += 4) {
    v2f a = *(const v2f*)(arow + k0);
    const float* wb = W + (size_t)(k0 + half * 2) * DIM + l15;
    v2f b0, b1;
    b0.x = wb[0];   b0.y = wb[DIM];            // K+0, K+1 rows of this half
    b1.x = wb[16];  b1.y = wb[DIM + 16];       // second N tile
    c0 = __builtin_amdgcn_wmma_f32_16x16x4_f32(false, a, false, b0, (short)0, c0, false, false);
    c1 = __builtin_amdgcn_wmma_f32_16x16x4_f32(false, a, false, b1, (short)0, c1, false, false);
  }

  float* yrow = Y + (size_t)(m0 + half * 8) * DIM + l15;
  #pragma unroll
  for (int v = 0; v < 8; ++v) {
    yrow[(size_t)v * DIM]      = c0[v];
    yrow[(size_t)v * DIM + 16] = c1[v];
  }
}

// ---------------------------------------------------------------------------
// Kernel 2: edge scatter-add  agg[dst] += y[src]  (32 floats per edge)
// One thread = one edge x one float4 chunk -> global_atomic_add_f32
// ---------------------------------------------------------------------------
__global__ __launch_bounds__(256)
void edge_agg(const float* __restrict__ Y, const int* __restrict__ src,
              const int* __restrict__ dst, float* __restrict__ AGG) {
  const int tid = blockIdx.x * blockDim.x + threadIdx.x;
  if (tid >= N_EDGES * 8) return;
  const int e = tid >> 3;
  const int q = (tid & 7) * 4;
  const int s = src[e];
  const int d = dst[e];
  v4f v = *(const v4f*)(Y + (size_t)s * DIM + q);
  float* ap = AGG + (size_t)d * DIM + q;
  atomicAdd(ap + 0, v.x);
  atomicAdd(ap + 1, v.y);
  atomicAdd(ap + 2, v.z);
  atomicAdd(ap + 3, v.w);
}

// ---------------------------------------------------------------------------
// Kernel 3: Hout = BN( relu( relu(Y + AGG + b1) @ W2 + b2 ) )
// A fragment built on the fly from Y+AGG+b1 with ReLU; BN fused in epilogue.
// ---------------------------------------------------------------------------
__global__ __launch_bounds__(32 * WAVES_PER_BLOCK)
void gin_lin2_wmma(const float* __restrict__ Y, const float* __restrict__ AGG,
                   const float* __restrict__ b1,
                   const float* __restrict__ W2, const float* __restrict__ b2,
                   const float* __restrict__ gamma, const float* __restrict__ beta,
                   const float* __restrict__ mean,  const float* __restrict__ var,
                   float* __restrict__ Hout) {
  const int wave = threadIdx.x >> 5;
  const int tile = blockIdx.x * WAVES_PER_BLOCK + wave;
  if (tile >= M_TILES) return;
  const int lane = threadIdx.x & 31;
  const int half = lane >> 4;
  const int l15  = lane & 15;
  const int m0   = tile * 16;

  v8f c0 = {}; v8f c1 = {};
  const size_t rbase = (size_t)(m0 + l15) * DIM + half * 2;

  #pragma unroll
  for (int k0 = 0; k0 < DIM; k0 += 4) {
    const int k = k0 + half * 2;
    v2f ya = *(const v2f*)(Y + rbase + k0);
    v2f ag = *(const v2f*)(AGG + rbase + k0);
    v2f a;
    a.x = fmaxf(ya.x + ag.x + b1[k],     0.0f);
    a.y = fmaxf(ya.y + ag.y + b1[k + 1], 0.0f);
    const float* wb = W2 + (size_t)k * DIM + l15;
    v2f b0, bb1;
    b0.x  = wb[0];   b0.y  = wb[DIM];
    bb1.x = wb[16];  bb1.y = wb[DIM + 16];
    c0 = __builtin_amdgcn_wmma_f32_16x16x4_f32(false, a, false, b0,  (short)0, c0, false, false);
    c1 = __builtin_amdgcn_wmma_f32_16x16x4_f32(false, a, false, bb1, (short)0, c1, false, false);
  }

  // epilogue: +b2, relu, batchnorm (eval)
  const int n0 = l15, n1 = l15 + 16;
  const float s0 = rsqrtf(var[n0] + BN_EPS) * gamma[n0];
  const float s1 = rsqrtf(var[n1] + BN_EPS) * gamma[n1];
  const float t0 = beta[n0] - mean[n0] * s0;
  const float t1 = beta[n1] - mean[n1] * s1;
  const float z0 = b2[n0], z1 = b2[n1];

  float* hrow = Hout + (size_t)(m0 + half * 8) * DIM + l15;
  #pragma unroll
  for (int v = 0; v < 8; ++v) {
    hrow[(size_t)v * DIM]      = fmaxf(c0[v] + z0, 0.0f) * s0 + t0;
    hrow[(size_t)v * DIM + 16] = fmaxf(c1[v] + z1, 0.0f) * s1 + t1;
  }
}

// ---------------------------------------------------------------------------
// Kernel 4: global_add_pool  G[batch[n]] += H[n]
// ---------------------------------------------------------------------------
__global__ __launch_bounds__(256)
void pool_kernel(const float* __restrict__ H, const int* __restrict__ batch,
                 float* __restrict__ G) {
  const int tid = blockIdx.x * blockDim.x + threadIdx.x;
  if (tid >= N_NODES * 8) return;
  const int n = tid >> 3;
  const int q = (tid & 7) * 4;
  const int b = batch[n];
  v4f v = *(const v4f*)(H + (size_t)n * DIM + q);
  float* gp = G + (size_t)b * DIM + q;
  atomicAdd(gp + 0, v.x);
  atomicAdd(gp + 1, v.y);
  atomicAdd(gp + 2, v.z);
  atomicAdd(gp + 3, v.w);
}

// ---------------------------------------------------------------------------
// Kernel 5: head: relu(G@fc1W+fc1b) @ fc2W + fc2b -> log_softmax
// ---------------------------------------------------------------------------
__global__ __launch_bounds__(128)
void head_kernel(const float* __restrict__ G,
                 const float* __restrict__ W1, const float* __restrict__ b1,
                 const float* __restrict__ W2, const float* __restrict__ b2,
                 float* __restrict__ out) {
  const int t = blockIdx.x * blockDim.x + threadIdx.x;
  if (t >= N_GRAPHS) return;
  float g[DIM];
  #pragma unroll
  for (int i = 0; i < DIM; ++i) g[i] = G[(size_t)t * DIM + i];
  float l0 = b2[0], l1 = b2[1];
  #pragma unroll 4
  for (int j = 0; j < DIM; ++j) {
    float s = b1[j];
    #pragma unroll
    for (int k = 0; k < DIM; ++k) s = fmaf(g[k], W1[k * DIM + j], s);
    s = fmaxf(s, 0.0f);
    l0 = fmaf(s, W2[j * N_CLASSES + 0], l0);
    l1 = fmaf(s, W2[j * N_CLASSES + 1], l1);
  }
  const float m = fmaxf(l0, l1);
  const float lse = m + logf(expf(l0 - m) + expf(l1 - m));
  out[t * 2 + 0] = l0 - lse;
  out[t * 2 + 1] = l1 - lse;
}

// ---------------------------------------------------------------------------
// Host launch
// Input order (setup_inputs dict order, params in insertion order):
//  0 x | 1 edge_index | 2 batch
//  3 + 4*L : convL.lin1.W, lin1.b, lin2.W, lin2.b   (L = 0..4)
// 23 + 4*L : bnL.gamma, beta, mean, var             (L = 0..4)
// 43 fc1.W | 44 fc1.b | 45 fc2.W | 46 fc2.b
// ---------------------------------------------------------------------------
extern "C" void kernel_launch(void* const* d_in, const int* in_sizes, int n_in,
                              void* d_out, int out_size, void* d_ws, size_t ws_size,
                              hipStream_t stream) {
  const float* x     = (const float*)d_in[0];
  const int*   ei    = (const int*)d_in[1];
  const int*   batch = (const int*)d_in[2];
  const int* src = ei;
  const int* dst = ei + N_EDGES;

  // workspace layout
  float* y    = (float*)d_ws;                       // N_NODES*DIM
  float* agg  = y   + (size_t)N_NODES * DIM;        // N_NODES*DIM
  float* hA   = agg + (size_t)N_NODES * DIM;        // N_NODES*DIM
  float* hB   = hA  + (size_t)N_NODES * DIM;        // N_NODES*DIM
  float* g    = hB  + (size_t)N_NODES * DIM;        // N_GRAPHS*DIM

  const int gemm_grid = (M_TILES + WAVES_PER_BLOCK - 1) / WAVES_PER_BLOCK;
  const int gemm_blk  = 32 * WAVES_PER_BLOCK;
  const int edge_grid = (N_EDGES * 8 + 255) / 256;
  const int pool_grid = (N_NODES * 8 + 255) / 256;

  const float* hin = x;
  int din = N_FEAT;
  float* hout = hA;

  for (int L = 0; L < 5; ++L) {
    const float* W1 = (const float*)d_in[3 + 4 * L + 0];
    const float* b1 = (const float*)d_in[3 + 4 * L + 1];
    const float* W2 = (const float*)d_in[3 + 4 * L + 2];
    const float* b2 = (const float*)d_in[3 + 4 * L + 3];
    const float* gm = (const float*)d_in[23 + 4 * L + 0];
    const float* bt = (const float*)d_in[23 + 4 * L + 1];
    const float* mn = (const float*)d_in[23 + 4 * L + 2];
    const float* vr = (const float*)d_in[23 + 4 * L + 3];

    // y = hin @ W1   (linearity: (h + A h) W1 == y + A y, y = h W1)
    gin_lin1_wmma<<<gemm_grid, gemm_blk, 0, stream>>>(hin, W1, y, din);
    // agg = segment_sum(y[src], dst)
    hipMemsetAsync(agg, 0, (size_t)N_NODES * DIM * sizeof(float), stream);
    edge_agg<<<edge_grid, 256, 0, stream>>>(y, src, dst, agg);
    // hout = bn(relu(relu(y + agg + b1) @ W2 + b2))
    gin_lin2_wmma<<<gemm_grid, gemm_blk, 0, stream>>>(y, agg, b1, W2, b2,
                                                      gm, bt, mn, vr, hout);
    hin = hout;
    hout = (hout == hA) ? hB : hA;
    din = DIM;
  }

  // global_add_pool
  hipMemsetAsync(g, 0, (size_t)N_GRAPHS * DIM * sizeof(float), stream);
  pool_kernel<<<pool_grid, 256, 0, stream>>>(hin, batch, g);

  // head
  const float* fc1W = (const float*)d_in[43];
  const float* fc1b = (const float*)d_in[44];
  const float* fc2W = (const float*)d_in[45];
  const float* fc2b = (const float*)d_in[46];
  head_kernel<<<(N_GRAPHS + 127) / 128, 128, 0, stream>>>(g, fc1W, fc1b, fc2W, fc2b,
                                                          (float*)d_out);
}